// get_loss_37701222924974
// MI455X (gfx1250) — compile-verified
//
#include <hip/hip_runtime.h>
#include <hip/hip_bf16.h>
#include <math.h>

#define H_IMG 375
#define W_IMG 1242
#define BATCH 8
#define NPTS  122880
#define BLOCKS_PER_B (NPTS / 256)   // 480

// workspace layout (in floats)
#define WS_A    0                    // B x 2 x 16 x 4 composite matrices (A_lo, A_hi) = 1024
#define WS_ROT  1024                 // rotationLoss[b] (8)
#define WS_BLK  1032                 // per-block manhattan partial sums (B * 480)

typedef __attribute__((ext_vector_type(2))) float v2f;
typedef __attribute__((ext_vector_type(8))) float v8f;

// ---------------------------------------------------------------------------
// Setup: per-batch composite 16x4 matrices (two variants) + rotation loss.
// Composite rows (each applied to raw homogeneous point [x,y,z,1]):
//   row 0-2 : (targetTransform@M - predTransform@M)   -> diff.xyz (for L1)
//   row 3-5 : (P_rect @ transOnly) @ M                -> proj.xyz
//   row 6   : [0,0,1,tz] @ M                          -> zc
// A_lo holds them in matrix rows 0-6  (D rows 0-7 -> lanes 0-15)
// A_hi holds them in matrix rows 8-14 (D rows 8-15 -> lanes 16-31)
// M = R_rect @ RT.
// ---------------------------------------------------------------------------
__global__ void setup_kernel(const float* __restrict__ predT,
                             const float* __restrict__ tgt,
                             const float* __restrict__ P,
                             const float* __restrict__ Rr,
                             const float* __restrict__ RT,
                             float* __restrict__ ws)
{
    int b = threadIdx.x;
    if (b >= BATCH) return;

    float M[4][4];
    for (int i = 0; i < 4; ++i)
        for (int j = 0; j < 4; ++j) {
            float s = 0.f;
            for (int k = 0; k < 4; ++k) s += Rr[i * 4 + k] * RT[k * 4 + j];
            M[i][j] = s;
        }

    float qw = predT[b * 7 + 0], qx = predT[b * 7 + 1];
    float qy = predT[b * 7 + 2], qz = predT[b * 7 + 3];
    float t0 = predT[b * 7 + 4], t1 = predT[b * 7 + 5], t2 = predT[b * 7 + 6];
    float qn = sqrtf(qw * qw + qx * qx + qy * qy + qz * qz);
    qw /= qn; qx /= qn; qy /= qn; qz /= qn;

    float R[3][3];
    R[0][0] = 1.f - 2.f * (qy * qy + qz * qz);
    R[0][1] = 2.f * (qx * qy - qw * qz);
    R[0][2] = 2.f * (qx * qz + qw * qy);
    R[1][0] = 2.f * (qx * qy + qw * qz);
    R[1][1] = 1.f - 2.f * (qx * qx + qz * qz);
    R[1][2] = 2.f * (qy * qz - qw * qx);
    R[2][0] = 2.f * (qx * qz - qw * qy);
    R[2][1] = 2.f * (qy * qz + qw * qx);
    R[2][2] = 1.f - 2.f * (qx * qx + qy * qy);

    // rotationLoss = || R^T @ gtR - I ||_F
    float rl = 0.f;
    for (int i = 0; i < 3; ++i)
        for (int j = 0; j < 3; ++j) {
            float e = 0.f;
            for (int k = 0; k < 3; ++k) e += R[k][i] * tgt[b * 16 + k * 4 + j];
            e -= (i == j) ? 1.f : 0.f;
            rl += e * e;
        }
    ws[WS_ROT + b] = sqrtf(rl);

    float C[7][4];                                    // composite rows
    float tv[3] = {t0, t1, t2};
    for (int i = 0; i < 3; ++i)                       // diff rows: (T - Pred) @ M
        for (int j = 0; j < 4; ++j) {
            float sT = 0.f;
            for (int k = 0; k < 4; ++k) sT += tgt[b * 16 + i * 4 + k] * M[k][j];
            float sP = tv[i] * M[3][j];
            for (int k = 0; k < 3; ++k) sP += R[i][k] * M[k][j];
            C[i][j] = sT - sP;
        }
    for (int i = 0; i < 3; ++i) {                     // proj rows: (P_rect@transOnly)@M
        float p3 = P[i * 4 + 0] * t0 + P[i * 4 + 1] * t1 + P[i * 4 + 2] * t2 + P[i * 4 + 3];
        for (int j = 0; j < 4; ++j) {
            float s = p3 * M[3][j];
            for (int k = 0; k < 3; ++k) s += P[i * 4 + k] * M[k][j];
            C[3 + i][j] = s;
        }
    }
    for (int j = 0; j < 4; ++j)                       // zc row
        C[6][j] = M[2][j] + t2 * M[3][j];

    float* Alo = ws + WS_A + b * 128;
    float* Ahi = Alo + 64;
    for (int r = 0; r < 16; ++r)
        for (int c = 0; c < 4; ++c) {
            Alo[r * 4 + c] = (r < 7) ? C[r][c] : 0.f;
            Ahi[r * 4 + c] = (r >= 8 && r < 15) ? C[r - 8][c] : 0.f;
        }
}

// Zero the rasterize target planes (channels 0 and 2 of the output image).
__global__ void clear_kernel(float* __restrict__ img)
{
    const long HW = (long)H_IMG * W_IMG;
    long i = (long)blockIdx.x * blockDim.x + threadIdx.x;
    const long total = (long)BATCH * 2 * HW;
    if (i >= total) return;
    long b = i / (2 * HW);
    long r = i % (2 * HW);
    long ch = (r < HW) ? 0 : 2;
    long p  = (r < HW) ? r : r - HW;
    img[b * 3 * HW + ch * HW + p] = 0.f;
}

// ---------------------------------------------------------------------------
// Main point pass: each wave processes 32 points with 2x v_wmma_f32_16x16x4_f32.
// A (16x4, f32): lane m<16 holds K=0,1; lane m+16 holds K=2,3 (ISA 7.12.2).
// B (4x16):      lane n<16 holds B[0][n],B[1][n]; lane n+16 holds B[2][n],B[3][n].
// D (16x16):     VGPR j: lanes 0-15 -> D[j][n], lanes 16-31 -> D[j+8][n].
// WMMA1 = A_lo x B(points 0..15)  -> lanes 0-15 consume their own point.
// WMMA2 = A_hi x B(points 16..31) -> lanes 16-31 consume their own point.
// Zero cross-lane traffic on results; only 3 shuffles to build B operands.
// ---------------------------------------------------------------------------
__global__ void points_kernel(const float* __restrict__ ptCld,
                              const int* __restrict__ ptSize,
                              const float* __restrict__ ws,
                              float* __restrict__ img,
                              float* __restrict__ blkSums)
{
    const int b    = blockIdx.y;
    const int lane = threadIdx.x & 31;
    const int wave = threadIdx.x >> 5;
    const int base = blockIdx.x * 256 + wave * 32;
    const long HW  = (long)H_IMG * W_IMG;

    __shared__ float s_w[8];

    // Composite A matrices -> 2+2 VGPRs per lane
    const int m  = lane & 15;
    const int kh = (lane < 16) ? 0 : 2;
    const float* Ab = ws + WS_A + b * 128 + m * 4 + kh;
    v2f alo, ahi;
    alo.x = Ab[0];
    alo.y = Ab[1];
    ahi.x = Ab[64];
    ahi.y = Ab[65];

    // Coalesced 128-bit load: one point per lane (grid covers N exactly)
    const int p = base + lane;
    const float4* pc = (const float4*)ptCld + (long)b * NPTS;
    float4 pt = pc[p];

    // Speculative prefetch of the next block's chunk (dropped safely at end)
    __builtin_prefetch((const void*)(pc + p + 256), 0, 1);

    // B operand for points base..base+15: lanes<16 need own x,y; lanes>=16 need z of lane-16
    float zlo = __shfl(pt.z, lane & 15);
    v2f bm1;
    bm1.x = (lane < 16) ? pt.x : zlo;
    bm1.y = (lane < 16) ? pt.y : 1.0f;

    // B operand for points base+16..base+31: lanes<16 need x,y of lane+16; lanes>=16 own z
    float xhi = __shfl(pt.x, (lane & 15) + 16);
    float yhi = __shfl(pt.y, (lane & 15) + 16);
    v2f bm2;
    bm2.x = (lane < 16) ? xhi : pt.z;
    bm2.y = (lane < 16) ? yhi : 1.0f;

    v8f c0 = {};
    v8f d1 = __builtin_amdgcn_wmma_f32_16x16x4_f32(false, alo, false, bm1,
                                                   (short)0, c0, false, false);
    v8f d2 = __builtin_amdgcn_wmma_f32_16x16x4_f32(false, ahi, false, bm2,
                                                   (short)0, c0, false, false);

    // Each lane's own point results: lanes<16 from d1 rows 0-6, lanes>=16 from d2 rows 8-14
    float r[7];
#pragma unroll
    for (int j = 0; j < 7; ++j)
        r[j] = (lane < 16) ? d1[j] : d2[j];

    const int sz = ptSize[b];
    const bool valid = (p < sz);

    // Manhattan L1 from pre-subtracted composite rows
    float l1  = fabsf(r[0]) + fabsf(r[1]) + fabsf(r[2]);
    float l1v = valid ? l1 : 0.f;

    // Projection + scatter-max rasterization
    float u  = r[3] / r[5];
    float v  = r[4] / r[5];
    float zc = r[6];
    bool mask = valid && (u >= 0.f) && (u < (float)W_IMG) &&
                (v >= 0.f) && (v < (float)H_IMG) && (zc > 0.f);
    if (mask) {
        int ui = min(max((int)u, 0), W_IMG - 1);
        int vi = min(max((int)v, 0), H_IMG - 1);
        long idx = (long)vi * W_IMG + ui;
        unsigned int* dptr = (unsigned int*)(img + (long)b * 3 * HW + idx);
        unsigned int* iptr = (unsigned int*)(img + (long)b * 3 * HW + 2 * HW + idx);
        atomicMax(dptr, __float_as_uint(zc));      // values >= 0: uint order == float order
        atomicMax(iptr, __float_as_uint(pt.w));
    }

    // Deterministic reduction: fixed-order shuffle tree + fixed-order LDS sum
#pragma unroll
    for (int off = 16; off > 0; off >>= 1)
        l1v += __shfl_down(l1v, off);
    if (lane == 0) s_w[wave] = l1v;
    __syncthreads();
    if (threadIdx.x == 0) {
        float s = 0.f;
        for (int w = 0; w < 8; ++w) s += s_w[w];
        blkSums[(long)b * BLOCKS_PER_B + blockIdx.x] = s;
    }
}

// In-place image normalization: ch0/ch1 from raw depth, ch2 from raw intensity.
__global__ void finalize_img(float* __restrict__ img)
{
    const long HW = (long)H_IMG * W_IMG;
    long i = (long)blockIdx.x * blockDim.x + threadIdx.x;
    if (i >= (long)BATCH * HW) return;
    long b = i / HW, p = i % HW;
    float* base = img + b * 3 * HW;
    float d  = base[p];
    float it = base[2 * HW + p];
    base[p]          = (d  - 0.485f) / 0.229f;
    base[HW + p]     = (d  - 0.456f) / 0.224f;
    base[2 * HW + p] = (it - 0.406f) / 0.225f;
}

// Losses: fixed-order summation over per-block partials -> deterministic.
__global__ void finalize_loss(const float* __restrict__ ws,
                              const int* __restrict__ ptSize,
                              float* __restrict__ out)
{
    if (threadIdx.x != 0 || blockIdx.x != 0) return;
    float mean = 0.f;
    for (int b = 0; b < BATCH; ++b) {
        float s = 0.f;
        for (int i = 0; i < BLOCKS_PER_B; ++i)
            s += ws[WS_BLK + (long)b * BLOCKS_PER_B + i];
        mean += s / (float)ptSize[b];
    }
    mean /= (float)BATCH;
    out[8] = mean;
    for (int b = 0; b < BATCH; ++b)
        out[b] = mean + ws[WS_ROT + b];
}

extern "C" void kernel_launch(void* const* d_in, const int* in_sizes, int n_in,
                              void* d_out, int out_size, void* d_ws, size_t ws_size,
                              hipStream_t stream)
{
    const float* predT  = (const float*)d_in[0];
    // d_in[1] srcClrT, d_in[2] srcDepthT: unused by the reference outputs
    const float* ptCld  = (const float*)d_in[3];
    const int*   ptSize = (const int*)d_in[4];
    const float* tgt    = (const float*)d_in[5];
    const float* P_rect = (const float*)d_in[6];
    const float* R_rect = (const float*)d_in[7];
    const float* RT     = (const float*)d_in[8];

    float* out = (float*)d_out;
    float* img = out + 9;                 // (B,3,H,W) image region
    float* ws  = (float*)d_ws;

    setup_kernel<<<1, 32, 0, stream>>>(predT, tgt, P_rect, R_rect, RT, ws);

    const long HW = (long)H_IMG * W_IMG;
    long clearN = (long)BATCH * 2 * HW;
    clear_kernel<<<(int)((clearN + 255) / 256), 256, 0, stream>>>(img);

    dim3 grid(BLOCKS_PER_B, BATCH);
    points_kernel<<<grid, 256, 0, stream>>>(ptCld, ptSize, ws, img, ws + WS_BLK);

    long fn = (long)BATCH * HW;
    finalize_img<<<(int)((fn + 255) / 256), 256, 0, stream>>>(img);

    finalize_loss<<<1, 32, 0, stream>>>(ws, ptSize, out);
}